// SpatialSelfAttention_33105607918177
// MI455X (gfx1250) — compile-verified
//
#include <hip/hip_runtime.h>
#include <hip/hip_bf16.h>

typedef float v2f __attribute__((ext_vector_type(2)));
typedef float v8f __attribute__((ext_vector_type(8)));
typedef unsigned int u32x4 __attribute__((ext_vector_type(4)));
typedef int i32x4 __attribute__((ext_vector_type(4)));
typedef int i32x8 __attribute__((ext_vector_type(8)));

#define C_DIM 512
#define N_DIM 1024
#define NHEAD 8
#define HDIM  64
#define BATCH 8
#define GNG   8
#define GNCH  64

#if defined(__HIP_DEVICE_COMPILE__) && __has_builtin(__builtin_amdgcn_tensor_load_to_lds)
#define HAVE_TDM 1
#else
#define HAVE_TDM 0
#endif

__device__ __forceinline__ v8f wmma4(v2f a, v2f b, v8f c) {
  return __builtin_amdgcn_wmma_f32_16x16x4_f32(
      false, a, false, b, (short)0, c, false, false);
}

__device__ __forceinline__ unsigned lds_off(const void* p) {
  return (unsigned)(uintptr_t)p;   // generic LDS pointer: low 32 bits = LDS offset
}

// Issue one TDM 2D tile load global->LDS (wave-level op; EXEC ignored).
// tile_w contiguous elements per row, tile_h rows, row stride in elements.
// Optional LDS padding: pad_amt_enc+1... per ISA: interval enc v -> 2^(v+1) dwords,
// amount enc v -> v+1 dwords inserted.
__device__ __forceinline__ void tdm_load_tile(const float* g, unsigned lds_byte_off,
                                              int tile_w, int tile_h,
                                              long long row_stride_elems,
                                              int pad_int_enc, int pad_amt_enc,
                                              int pad_en) {
#if HAVE_TDM
  unsigned long long ga = (unsigned long long)(uintptr_t)g;
  u32x4 g0 = { 1u,                                   // count=1, user mode
               lds_byte_off,                         // lds_addr
               (unsigned)ga,                         // global_addr[31:0]
               (unsigned)((ga >> 32) & 0x01FFFFFFull) | (2u << 30) }; // [56:32] | type=2
  unsigned long long s0 = (unsigned long long)row_stride_elems;
  unsigned d0 = (2u << 16);                          // data_size = 4B
  if (pad_en)
    d0 |= (1u << 20) | ((unsigned)pad_int_enc << 22) | ((unsigned)pad_amt_enc << 25);
  const unsigned td0 = 1u << 20, td1 = 1u << 20;     // generous tensor dims (no OOB)
  i32x8 g1 = {
      (int)d0,
      (int)((td0 & 0xFFFFu) << 16),                               // tensor_dim0 lo
      (int)(((td0 >> 16) & 0xFFFFu) | ((td1 & 0xFFFFu) << 16)),   // dim0 hi | dim1 lo
      (int)(((td1 >> 16) & 0xFFFFu) | (((unsigned)tile_w) << 16)),// dim1 hi | tile_dim0
      (int)((unsigned)tile_h & 0xFFFFu),                          // tile_dim1 (tile_dim2=0)
      (int)(unsigned)(s0 & 0xFFFFFFFFull),                        // dim0_stride[31:0]
      (int)(unsigned)((s0 >> 32) & 0xFFFFull),                    // stride hi | dim1_stride=0
      0 };
  i32x4 z4 = {0, 0, 0, 0};
#if __clang_major__ >= 23
  i32x8 z8 = {0, 0, 0, 0, 0, 0, 0, 0};
  __builtin_amdgcn_tensor_load_to_lds(g0, g1, z4, z4, z8, 0);
#else
  __builtin_amdgcn_tensor_load_to_lds(g0, g1, z4, z4, 0);
#endif
#else
  (void)g; (void)lds_byte_off; (void)tile_w; (void)tile_h;
  (void)row_stride_elems; (void)pad_int_enc; (void)pad_amt_enc; (void)pad_en;
#endif
}

// -------------------------------------------------------------------------
// GEMM: Out[b][o][n] = sum_c W[o][c] * X[b][c][n]
// block = 256 threads (8 waves); tile = 128(o) x 64(n); K chunks of 64.
// grid = (N/64, O/128, B)
// -------------------------------------------------------------------------
__global__ __launch_bounds__(256)
void gemm_f32_wmma(const float* __restrict__ W, const float* __restrict__ X,
                   float* __restrict__ Out, int Ochan) {
  __shared__ __align__(16) float xs[64 * 64];   // [k][n], stride 64
  __shared__ __align__(16) float wl[128 * 66];  // [o][k], stride 66 (TDM row pad)

  const int tid  = threadIdx.x;
  const int wave = tid >> 5;
  const int lane = tid & 31;
  const int lo   = lane & 15;
  const int hi   = lane >> 4;

  const int n0 = blockIdx.x * 64;
  const int o0 = blockIdx.y * 128;
  const int b  = blockIdx.z;

  v8f acc[4] = { {}, {}, {}, {} };

  for (int k0 = 0; k0 < C_DIM; k0 += 64) {
    __syncthreads();   // previous tile fully consumed
#if HAVE_TDM
    if (wave == 0) {
      tdm_load_tile(&X[((size_t)b * C_DIM + k0) * N_DIM + n0], lds_off(xs),
                    64, 64, N_DIM, 0, 0, 0);
      tdm_load_tile(&W[(size_t)o0 * C_DIM + k0], lds_off(wl),
                    64, 128, C_DIM, /*interval 64dw*/5, /*amount 2dw*/1, 1);
      __builtin_amdgcn_s_wait_tensorcnt(0);
    }
#else
    for (int i = tid; i < 64 * 64; i += 256) {
      int kk = i >> 6, nn = i & 63;
      xs[kk * 64 + nn] = X[((size_t)b * C_DIM + (k0 + kk)) * N_DIM + n0 + nn];
    }
    for (int i = tid; i < 128 * 64; i += 256) {
      int oo = i >> 6, kk = i & 63;
      wl[oo * 66 + kk] = W[(size_t)(o0 + oo) * C_DIM + k0 + kk];
    }
#endif
    __syncthreads();

#pragma unroll
    for (int kk = 0; kk < 64; kk += 4) {
      v2f a = *(const v2f*)&wl[(16 * wave + lo) * 66 + kk + 2 * hi];
#pragma unroll
      for (int t = 0; t < 4; ++t) {
        v2f bb;
        bb.x = xs[(kk + 2 * hi) * 64 + t * 16 + lo];
        bb.y = xs[(kk + 1 + 2 * hi) * 64 + t * 16 + lo];
        acc[t] = wmma4(a, bb, acc[t]);
      }
    }
  }

#pragma unroll
  for (int t = 0; t < 4; ++t)
#pragma unroll
    for (int v = 0; v < 8; ++v) {
      int o = o0 + 16 * wave + v + 8 * hi;
      Out[((size_t)b * Ochan + o) * N_DIM + n0 + t * 16 + lo] = acc[t][v];
    }
}

// -------------------------------------------------------------------------
// Flash attention per (b, h): q,k,v are [64 d][1024 n] slabs inside qkv.
// block = 256 threads (8 waves); 128 queries per block; keys in blocks of 32.
// grid = (N/128, NHEAD, B)
// -------------------------------------------------------------------------
__global__ __launch_bounds__(256)
void flash_attn_wmma(const float* __restrict__ qkv, float* __restrict__ aout) {
  __shared__ __align__(16) float ks[64 * 32];      // [d][m], stride 32
  __shared__ __align__(16) float vs[64 * 34];      // [d][m], stride 34 (TDM row pad)
  __shared__ __align__(16) float pt[32][129];      // P^T: [m][128 n] (+1 pad)

  const int tid  = threadIdx.x;
  const int wave = tid >> 5;
  const int lane = tid & 31;
  const int lo   = lane & 15;
  const int hi   = lane >> 4;

  const int nblk = blockIdx.x * 128;
  const int h    = blockIdx.y;
  const int b    = blockIdx.z;

  const float* Q = qkv + ((size_t)b * 3 * C_DIM + h * HDIM) * N_DIM;
  const float* K = Q + (size_t)C_DIM * N_DIM;
  const float* V = K + (size_t)C_DIM * N_DIM;

  // this wave's q strip, preloaded in WMMA A-layout
  const int nq = nblk + 16 * wave + lo;
  float qreg[32];
#pragma unroll
  for (int kk4 = 0; kk4 < 16; ++kk4) {
    qreg[2 * kk4]     = Q[(size_t)(kk4 * 4 + 2 * hi)     * N_DIM + nq];
    qreg[2 * kk4 + 1] = Q[(size_t)(kk4 * 4 + 1 + 2 * hi) * N_DIM + nq];
  }

  const float scale = 0.125f;  // hd^-0.5
  float mrow[8], lrow[8];
#pragma unroll
  for (int v = 0; v < 8; ++v) { mrow[v] = -1e30f; lrow[v] = 0.0f; }
  v8f o0 = {}, o1 = {}, o2 = {}, o3 = {};

  for (int mb = 0; mb < N_DIM / 32; ++mb) {
    const int mblk = mb * 32;
    __syncthreads();
#if HAVE_TDM
    if (wave == 0) {
      tdm_load_tile(&K[mblk], lds_off(ks), 32, 64, N_DIM, 0, 0, 0);
      tdm_load_tile(&V[mblk], lds_off(vs), 32, 64, N_DIM, /*interval 32dw*/4, /*amount 2dw*/1, 1);
      __builtin_amdgcn_s_wait_tensorcnt(0);
    }
#else
    for (int i = tid; i < 64 * 32; i += 256) {
      int d = i >> 5, m = i & 31;
      ks[d * 32 + m] = K[(size_t)d * N_DIM + mblk + m];
      vs[d * 34 + m] = V[(size_t)d * N_DIM + mblk + m];
    }
#endif
    __syncthreads();

    // S = q^T k : two 16x16 tiles covering 32 keys
    v8f s0 = {}, s1 = {};
#pragma unroll
    for (int kk4 = 0; kk4 < 16; ++kk4) {
      v2f a; a.x = qreg[2 * kk4]; a.y = qreg[2 * kk4 + 1];
      int kb = kk4 * 4 + 2 * hi;
      v2f b0; b0.x = ks[kb * 32 + lo];      b0.y = ks[(kb + 1) * 32 + lo];
      v2f b1; b1.x = ks[kb * 32 + 16 + lo]; b1.y = ks[(kb + 1) * 32 + 16 + lo];
      s0 = wmma4(a, b0, s0);
      s1 = wmma4(a, b1, s1);
    }

    // online softmax per row (row M = v + 8*hi lives in a 16-lane half)
#pragma unroll
    for (int v = 0; v < 8; ++v) {
      float e0 = s0[v] * scale, e1 = s1[v] * scale;
      float mx = fmaxf(e0, e1);
#pragma unroll
      for (int msk = 1; msk < 16; msk <<= 1)
        mx = fmaxf(mx, __shfl_xor(mx, msk, 32));
      float mnew  = fmaxf(mrow[v], mx);
      float alpha = __expf(mrow[v] - mnew);
      float p0 = __expf(e0 - mnew);
      float p1 = __expf(e1 - mnew);
      float rs = p0 + p1;
#pragma unroll
      for (int msk = 1; msk < 16; msk <<= 1)
        rs += __shfl_xor(rs, msk, 32);
      lrow[v] = lrow[v] * alpha + rs;
      mrow[v] = mnew;
      o0[v] *= alpha; o1[v] *= alpha; o2[v] *= alpha; o3[v] *= alpha;
      pt[lo][16 * wave + v + 8 * hi]      = p0;   // wave-private column slice
      pt[16 + lo][16 * wave + v + 8 * hi] = p1;
    }

    // O += P * V^T  (K = m, 32 wide)
#pragma unroll
    for (int kk4 = 0; kk4 < 8; ++kk4) {
      int kb = kk4 * 4 + 2 * hi;
      v2f a; a.x = pt[kb][16 * wave + lo]; a.y = pt[kb + 1][16 * wave + lo];
      v2f bb;
      bb = *(const v2f*)&vs[(lo)      * 34 + kb]; o0 = wmma4(a, bb, o0);
      bb = *(const v2f*)&vs[(16 + lo) * 34 + kb]; o1 = wmma4(a, bb, o1);
      bb = *(const v2f*)&vs[(32 + lo) * 34 + kb]; o2 = wmma4(a, bb, o2);
      bb = *(const v2f*)&vs[(48 + lo) * 34 + kb]; o3 = wmma4(a, bb, o3);
    }
  }

  // normalize and write out[b][h*64 + d][n]
#pragma unroll
  for (int v = 0; v < 8; ++v) {
    float inv = 1.0f / lrow[v];
    int n = nblk + 16 * wave + v + 8 * hi;
    aout[((size_t)b * C_DIM + h * HDIM +  0 + lo) * N_DIM + n] = o0[v] * inv;
    aout[((size_t)b * C_DIM + h * HDIM + 16 + lo) * N_DIM + n] = o1[v] * inv;
    aout[((size_t)b * C_DIM + h * HDIM + 32 + lo) * N_DIM + n] = o2[v] * inv;
    aout[((size_t)b * C_DIM + h * HDIM + 48 + lo) * N_DIM + n] = o3[v] * inv;
  }
}

// -------------------------------------------------------------------------
// GroupNorm stats: one block per (b,g); contiguous 64x1024 slab.
// -------------------------------------------------------------------------
__global__ __launch_bounds__(256)
void gn_stats(const float* __restrict__ proj, float* __restrict__ stats) {
  const int bg = blockIdx.x;
  const int b = bg >> 3, g = bg & 7;
  const float* base = proj + ((size_t)b * C_DIM + g * GNCH) * N_DIM;
  float s = 0.0f, s2 = 0.0f;
  for (int i = threadIdx.x; i < GNCH * N_DIM; i += 256) {
    float v = base[i];
    s += v; s2 += v * v;
  }
#pragma unroll
  for (int msk = 16; msk >= 1; msk >>= 1) {
    s  += __shfl_xor(s,  msk, 32);
    s2 += __shfl_xor(s2, msk, 32);
  }
  __shared__ float ss[8], ss2[8];
  const int wave = threadIdx.x >> 5;
  if ((threadIdx.x & 31) == 0) { ss[wave] = s; ss2[wave] = s2; }
  __syncthreads();
  if (threadIdx.x == 0) {
    float t = 0.0f, t2 = 0.0f;
#pragma unroll
    for (int w = 0; w < 8; ++w) { t += ss[w]; t2 += ss2[w]; }
    const float cnt = (float)(GNCH * N_DIM);
    float mean = t / cnt;
    stats[bg * 2]     = mean;
    stats[bg * 2 + 1] = t2 / cnt - mean * mean;
  }
}

// -------------------------------------------------------------------------
// Apply GN + affine + residual.
// -------------------------------------------------------------------------
__global__ __launch_bounds__(256)
void gn_apply(const float* __restrict__ proj, const float* __restrict__ stats,
              const float* __restrict__ gamma, const float* __restrict__ beta,
              const float* __restrict__ x, float* __restrict__ out) {
  size_t idx = (size_t)blockIdx.x * 256 + threadIdx.x;
  int c = (int)((idx >> 10) & (C_DIM - 1));
  int b = (int)(idx >> 19);
  int bg = b * GNG + (c >> 6);
  float mean = stats[bg * 2];
  float var  = stats[bg * 2 + 1];
  float r = rsqrtf(var + 1e-5f);
  out[idx] = (proj[idx] - mean) * r * gamma[c] + beta[c] + x[idx];
}

extern "C" void kernel_launch(void* const* d_in, const int* in_sizes, int n_in,
                              void* d_out, int out_size, void* d_ws, size_t ws_size,
                              hipStream_t stream) {
  const float* x      = (const float*)d_in[0];
  const float* w_qkv  = (const float*)d_in[1];
  const float* w_proj = (const float*)d_in[2];
  const float* gamma  = (const float*)d_in[3];
  const float* beta   = (const float*)d_in[4];
  float* out = (float*)d_out;

  float* qkv     = (float*)d_ws;                              // B*3C*N
  float* attnout = qkv + (size_t)BATCH * 3 * C_DIM * N_DIM;   // B*C*N
  float* proj    = attnout + (size_t)BATCH * C_DIM * N_DIM;   // B*C*N
  float* stats   = proj + (size_t)BATCH * C_DIM * N_DIM;      // B*G*2

  gemm_f32_wmma<<<dim3(N_DIM / 64, (3 * C_DIM) / 128, BATCH), 256, 0, stream>>>(
      w_qkv, x, qkv, 3 * C_DIM);
  flash_attn_wmma<<<dim3(N_DIM / 128, NHEAD, BATCH), 256, 0, stream>>>(qkv, attnout);
  gemm_f32_wmma<<<dim3(N_DIM / 64, C_DIM / 128, BATCH), 256, 0, stream>>>(
      w_proj, attnout, proj, C_DIM);
  gn_stats<<<BATCH * GNG, 256, 0, stream>>>(proj, stats);
  gn_apply<<<(BATCH * C_DIM * N_DIM) / 256, 256, 0, stream>>>(
      proj, stats, gamma, beta, x, out);
}